// GaussianDiffusion_67430986547317
// MI455X (gfx1250) — compile-verified
//
#include <hip/hip_runtime.h>

// GaussianDiffusion q_sample on MI455X (gfx1250):
// out[b,c] = C_b * X[b,c] * C_b, where C_b is the 128x128 symmetric circulant
// matrix of the composed 1D Gaussian kernel h_b = g_0 (*) ... (*) g_{t[b]}.
// Two fp32 WMMA GEMMs per image instead of up to 50 dependent blur sweeps.
// Staging uses GLOBAL_LOAD_ASYNC_TO_LDS (ASYNCcnt) to skip the VGPR round trip.

#define HW   128
#define LDP  132          // padded LDS row stride (floats): 528B rows (16B aligned),
                          // stride mod 64 banks = 4 -> conflict-free fragment loads

typedef __attribute__((ext_vector_type(2))) float v2f;
typedef __attribute__((ext_vector_type(8))) float v8f;

// ---------------------------------------------------------------------------
// Kernel 1: build composed 1D kernels h_b (fp64 for accuracy) and expand to
// circulant matrices Cmat[b][r][c] = h_b[(r - c) mod 128] in workspace.
// ---------------------------------------------------------------------------
__global__ void gd_build_circulant(const float* __restrict__ kernels,  // [50,3,3]
                                   const int*   __restrict__ t,        // [16]
                                   float*       __restrict__ Cmat) {   // [16,128,128]
    __shared__ double h[HW];
    const int b = blockIdx.x;
    const int j = threadIdx.x;                 // 0..127
    const int steps = t[b] + 1;                // all_blurs[t] == t+1 blur steps

    h[j] = (j == 0) ? 1.0 : 0.0;               // delta
    __syncthreads();

    for (int i = 0; i < steps; ++i) {
        // 1D factor of the separable 2D Gaussian: g[r] = row-sum of k
        const float* k9 = kernels + i * 9;
        double g0 = (double)k9[0] + (double)k9[1] + (double)k9[2];
        double g1 = (double)k9[3] + (double)k9[4] + (double)k9[5];
        double g2 = (double)k9[6] + (double)k9[7] + (double)k9[8];
        // circular 3-tap update: h'[m] = g0*h[m+1] + g1*h[m] + g2*h[m-1]
        double v = g0 * h[(j + 1) & (HW - 1)]
                 + g1 * h[j]
                 + g2 * h[(j - 1) & (HW - 1)];
        __syncthreads();
        h[j] = v;
        __syncthreads();
    }

    float* Cb = Cmat + (size_t)b * HW * HW;
    for (int r = 0; r < HW; ++r)
        Cb[r * HW + j] = (float)h[(r - j) & (HW - 1)];
}

// ---------------------------------------------------------------------------
// Kernel 2: per (b,c) image, Y = C * X * C with fp32 WMMA 16x16x4.
// 48 blocks x 256 threads (8 wave32s). LDS: X, C, T each 128x132 fp32 (~198KB).
// Each wave owns output column strip tn = wave*16 and walks the 8 row tiles,
// keeping the 32 B-fragments (full K strip) resident in VGPRs.
// ---------------------------------------------------------------------------
__global__ void gd_blur_gemm(const float* __restrict__ X,     // [16,3,128,128]
                             const float* __restrict__ Cmat,  // [16,128,128]
                             float*       __restrict__ Y) {   // [16,3,128,128]
    extern __shared__ float lds[];
    float* sX = lds;                    // [128][LDP]
    float* sC = lds + HW * LDP;         // [128][LDP]
    float* sT = lds + 2 * HW * LDP;     // [128][LDP]

    const int bc  = blockIdx.x;         // 0..47  (= b*3 + c)
    const int b   = bc / 3;
    const int tid = threadIdx.x;        // 0..255

    const float* gX = X    + (size_t)bc * HW * HW;
    const float* gC = Cmat + (size_t)b  * HW * HW;

    // ---- async-stage X and C into padded LDS (b128 per lane, ASYNCcnt) ----
    #pragma unroll 1
    for (int it = 0; it < 16; ++it) {
        int idx = tid + it * 256;            // 4096 16B transfers per matrix
        int row = idx >> 5;                  // 32 float4 per row
        int c4  = (idx & 31) << 2;
        unsigned goff = (unsigned)((row * HW + c4) * 4);       // global byte offset
        unsigned lX   = (unsigned)(size_t)&sX[row * LDP + c4]; // LDS byte offset
        unsigned lC   = (unsigned)(size_t)&sC[row * LDP + c4];
        asm volatile("global_load_async_to_lds_b128 %0, %1, %2"
                     :: "v"(lX), "v"(goff), "s"(gX) : "memory");
        asm volatile("global_load_async_to_lds_b128 %0, %1, %2"
                     :: "v"(lC), "v"(goff), "s"(gC) : "memory");
    }
    asm volatile("s_wait_asynccnt 0x0" ::: "memory");
    __syncthreads();

    const int lane  = tid & 31;
    const int wave  = tid >> 5;              // 0..7
    const int lrow  = lane & 15;
    const int khalf = (lane >> 4) << 1;      // lanes 0-15 -> K0,K1; lanes 16-31 -> K2,K3
    const int rhalf = (lane >> 4) << 3;      // D rows: p (lanes<16) / p+8 (lanes>=16)
    const int tn    = wave << 4;             // fixed output column strip per wave

    // ---- GEMM1: T = X * C ----
    {
        // Hoist the full K-strip of B (= C[:, tn+lrow]) into 32 v2f fragments
        v2f bfrag[32];
        #pragma unroll
        for (int kk = 0; kk < 32; ++kk) {
            bfrag[kk].x = sC[(4 * kk + khalf + 0) * LDP + tn + lrow];
            bfrag[kk].y = sC[(4 * kk + khalf + 1) * LDP + tn + lrow];
        }
        #pragma unroll 1
        for (int ti = 0; ti < 8; ++ti) {     // uniform trip count: no exec masking
            const int tm = ti << 4;
            v8f acc = {};
            #pragma unroll
            for (int kk = 0; kk < 32; ++kk) {
                v2f a = *(const v2f*)&sX[(tm + lrow) * LDP + 4 * kk + khalf];
                acc = __builtin_amdgcn_wmma_f32_16x16x4_f32(
                    false, a, false, bfrag[kk], (short)0, acc, false, false);
            }
            #pragma unroll
            for (int p = 0; p < 8; ++p)
                sT[(tm + rhalf + p) * LDP + tn + lrow] = acc[p];
        }
    }
    __syncthreads();

    // ---- GEMM2: Y = C * T, stream result straight to HBM ----
    float* gY = Y + (size_t)bc * HW * HW;
    {
        v2f bfrag[32];
        #pragma unroll
        for (int kk = 0; kk < 32; ++kk) {
            bfrag[kk].x = sT[(4 * kk + khalf + 0) * LDP + tn + lrow];
            bfrag[kk].y = sT[(4 * kk + khalf + 1) * LDP + tn + lrow];
        }
        #pragma unroll 1
        for (int ti = 0; ti < 8; ++ti) {
            const int tm = ti << 4;
            v8f acc = {};
            #pragma unroll
            for (int kk = 0; kk < 32; ++kk) {
                v2f a = *(const v2f*)&sC[(tm + lrow) * LDP + 4 * kk + khalf];
                acc = __builtin_amdgcn_wmma_f32_16x16x4_f32(
                    false, a, false, bfrag[kk], (short)0, acc, false, false);
            }
            #pragma unroll
            for (int p = 0; p < 8; ++p)
                __builtin_nontemporal_store(acc[p],
                    &gY[(tm + rhalf + p) * HW + tn + lrow]);
        }
    }
}

// ---------------------------------------------------------------------------
extern "C" void kernel_launch(void* const* d_in, const int* in_sizes, int n_in,
                              void* d_out, int out_size, void* d_ws, size_t ws_size,
                              hipStream_t stream) {
    const float* x       = (const float*)d_in[0];   // [16,3,128,128]
    const float* kernels = (const float*)d_in[1];   // [50,3,3]
    const int*   t       = (const int*)d_in[2];     // [16]
    float*       out     = (float*)d_out;           // [16,3,128,128]
    float*       Cmat    = (float*)d_ws;            // 16*128*128 fp32 = 1 MB

    gd_build_circulant<<<16, 128, 0, stream>>>(kernels, t, Cmat);

    const size_t ldsBytes = (size_t)3 * HW * LDP * sizeof(float);   // ~198 KB
    gd_blur_gemm<<<48, 256, ldsBytes, stream>>>(x, Cmat, out);
}